// BlockwiseQuantizationOptim_20572893347946
// MI455X (gfx1250) — compile-verified
//
#include <hip/hip_runtime.h>

// ---------------------------------------------------------------------------
// Blockwise soft-quantization for MI455X (gfx1250, wave32).
//   - TDM tensor_load_to_lds stages 128x64 f32 chunks of each 128x128 block.
//   - exp via v_exp_f32 (exp2 with folded -T*log2e scale, |src| modifier).
//   - einsum('nijl,nl->nij') numerator AND softmax denominator computed with
//     chained V_WMMA_F32_16X16X4_F32 (B cols 0..7 = bins, cols 8..15 = 1.0).
//   - A-operand half-exchange via v_permlane16_swap_b32 (1 op yields both
//     tiles' slots); ds_bpermute fallback if the builtin is unavailable.
//   - entropy: per-lane register bin_mass, block reduction -> d_ws, final sum
//     kernel -> d_out[R*C] (deterministic, no float atomics).
// ---------------------------------------------------------------------------

#define BSZ   128
#define RR    4096
#define CC    2048
#define NBCB  16          // blocks per row of blocks (C/BSZ)
#define NB    512
#define NBINS 31
#define EPSF  1e-6f

typedef float v2f  __attribute__((ext_vector_type(2)));
typedef float v8f  __attribute__((ext_vector_type(8)));
typedef unsigned int u32x4 __attribute__((ext_vector_type(4)));
typedef int   i32x4 __attribute__((ext_vector_type(4)));
typedef int   i32x8 __attribute__((ext_vector_type(8)));

// Build a D# for a 2D tile: tile_dim0 = 128 cols, tile_dim1 = nrows rows,
// element = 4 bytes, tensor row stride = CC elements. (6-arg clang-23 builtin.)
__device__ __forceinline__ void tdm_load_rows(const float* gsrc, unsigned lds_off,
                                              int nrows) {
    unsigned long long ga = (unsigned long long)(size_t)gsrc;
    u32x4 g0;
    g0[0] = 1u;                                        // count=1, user mode
    g0[1] = lds_off;                                   // lds_addr
    g0[2] = (unsigned)(ga & 0xFFFFFFFFu);              // global_addr[31:0]
    g0[3] = (unsigned)((ga >> 32) & 0x01FFFFFFu)       // global_addr[56:32]
          | (2u << 30);                                // type = 2 ("image")
    i32x8 g1;
    g1[0] = (int)(2u << 16);                           // data_size = 4B
    g1[1] = (int)(((unsigned)CC & 0xFFFFu) << 16);     // tensor_dim0 lo16
    g1[2] = (int)((((unsigned)CC >> 16) & 0xFFFFu)     // tensor_dim0 hi16
          | (((unsigned)RR & 0xFFFFu) << 16));         // tensor_dim1 lo16
    g1[3] = (int)((((unsigned)RR >> 16) & 0xFFFFu)     // tensor_dim1 hi16
          | ((unsigned)BSZ << 16));                    // tile_dim0 = 128
    g1[4] = nrows;                                     // tile_dim1 (tile_dim2=0)
    g1[5] = (int)CC;                                   // tensor_dim0_stride lo32
    g1[6] = 0;                                         // stride hi, dim1_stride lo
    g1[7] = 0;
    i32x4 z4 = {0, 0, 0, 0};                           // groups 2/3 unused (2D)
    i32x8 z8 = {0, 0, 0, 0, 0, 0, 0, 0};
    __builtin_amdgcn_tensor_load_to_lds(g0, g1, z4, z4, z8, 0);
}

__global__ __launch_bounds__(128)
void bq_main(const float* __restrict__ weight,
             const float* __restrict__ w_min,
             const float* __restrict__ w_max,
             const float* __restrict__ bins,
             float* __restrict__ out,
             float* __restrict__ ws) {
    __shared__ __align__(128) float tile[64 * BSZ];    // 32 KB staging chunk
    __shared__ float nd[4][64];                        // per-wave num/den bounce
    __shared__ float red[BSZ];

    const int n    = blockIdx.x;
    const int br   = n >> 4;                           // n / NBCB
    const int bc   = n & (NBCB - 1);
    const int tid  = threadIdx.x;
    const int wave = tid >> 5;
    const int lane = tid & 31;
    const int col  = tid;                              // column of the block owned

    // --- block scalars (uniform -> SGPRs) ---
    const float wmn    = w_min[n];
    const float wmx    = w_max[n];
    const float wmin_c = fminf(wmn, wmx - EPSF);
    const float wmax_c = fmaxf(wmx, wmin_c + EPSF);
    const float span   = wmax_c - wmin_c;
    const float inv    = 1.0f / (span + EPSF);

    float bv[NBINS];
#pragma unroll
    for (int k = 0; k < NBINS; ++k) bv[k] = bins[n * NBINS + k];

    // --- B-matrix lane constants for V_WMMA_F32_16X16X4_F32 ---
    // B is 4x16: VGPR0 -> K = 0(lanes 0-15) / 2(lanes 16-31); VGPR1 -> K+1.
    // Columns N<8 carry bins[k] (numerator), N>=8 carry 1.0 (denominator).
    const bool hiHalf  = lane >= 16;
    const bool colBins = (lane & 15) < 8;
    float B0g[8], B1g[8];
#pragma unroll
    for (int g = 0; g < 8; ++g) {
        int k0 = 4 * g + (hiHalf ? 2 : 0);
        int k1 = k0 + 1;
        float w0 = (k0 < NBINS) ? bv[k0] : 0.0f;
        float w1 = (k1 < NBINS) ? bv[k1] : 0.0f;
        B0g[g] = colBins ? w0 : 1.0f;
        B1g[g] = colBins ? w1 : 1.0f;
    }

    float mass[NBINS];
#pragma unroll
    for (int k = 0; k < NBINS; ++k) mass[k] = 0.0f;

    const float c2 = -144.26950408889634f;             // -T * log2(e), T = 100
    float* ndw  = nd[wave];
    float* orow = out + (size_t)(br * BSZ) * CC + bc * BSZ;
    float* tptr = tile;

    for (int chunk = 0; chunk < 2; ++chunk) {
        __syncthreads();                               // prior chunk fully read
        if (wave == 0) {
            const float* gsrc = weight
                + (size_t)(br * BSZ + chunk * 64) * CC + bc * BSZ;
            tdm_load_rows(gsrc, (unsigned)(size_t)(void*)tile, 64);
            __builtin_amdgcn_s_wait_tensorcnt(0);
        }
        __syncthreads();
        asm volatile("" : "+v"(tptr) :: "memory");     // TDM wrote LDS: no UB fold

        for (int ii = 0; ii < 64; ++ii) {
            const int i = chunk * 64 + ii;
            const float x  = tptr[ii * BSZ + col];
            const float wn = (x - wmin_c) * inv;

            // unnormalized softmax weights: e[k] = exp(-T*|wn-b_k|)
            float e[32];
#pragma unroll
            for (int k = 0; k < NBINS; ++k)
                e[k] = __builtin_amdgcn_exp2f(c2 * fabsf(wn - bv[k]));
            e[31] = 0.0f;                              // K=32 pad

            // Two 16-element tiles per wave; 8 chained K=4 WMMAs each.
            v8f c0 = {}; v8f c1 = {};
#pragma unroll
            for (int g = 0; g < 8; ++g) {
                v2f a0, a1, bb;
#if __has_builtin(__builtin_amdgcn_permlane16_swap)
                // vdst' = {own.lo, partner.lo}  -> tile0 A slot
                // vsrc' = {partner.hi, own.hi}  -> tile1 A slot
                {
                    auto r0 = __builtin_amdgcn_permlane16_swap(
                        __float_as_int(e[4 * g + 0]),
                        __float_as_int(e[4 * g + 2]), false, false);
                    a0[0] = __int_as_float(r0[0]);
                    a1[0] = __int_as_float(r0[1]);
                    auto r1 = __builtin_amdgcn_permlane16_swap(
                        __float_as_int(e[4 * g + 1]),
                        __float_as_int(e[4 * g + 3]), false, false);
                    a0[1] = __int_as_float(r1[0]);
                    a1[1] = __int_as_float(r1[1]);
                }
#else
                // half-exchange so both lane halves hold the A-layout slots
                {
                    float s0 = __shfl_xor(e[4 * g + 0], 16, 32);
                    float s1 = __shfl_xor(e[4 * g + 1], 16, 32);
                    float s2 = __shfl_xor(e[4 * g + 2], 16, 32);
                    float s3 = __shfl_xor(e[4 * g + 3], 16, 32);
                    a0[0] = hiHalf ? s2 : e[4 * g + 0];
                    a0[1] = hiHalf ? s3 : e[4 * g + 1];
                    a1[0] = hiHalf ? e[4 * g + 2] : s0;
                    a1[1] = hiHalf ? e[4 * g + 3] : s1;
                }
#endif
                bb[0] = B0g[g];
                bb[1] = B1g[g];
                c0 = __builtin_amdgcn_wmma_f32_16x16x4_f32(
                         false, a0, false, bb, (short)0, c0, false, false);
                c1 = __builtin_amdgcn_wmma_f32_16x16x4_f32(
                         false, a1, false, bb, (short)0, c1, false, false);
            }

            // D layout: VGPR v, lanes0-15 -> (M=v, N=lane); lanes16-31 -> M=v+8.
            // num[m] lives at N=0 (lanes 0,16), den[m] at N=8 (lanes 8,24).
            if ((lane & 7) == 0) {
                const int off0 = (lane == 0) ? 0 : (lane == 16) ? 8
                               : (lane == 8) ? 32 : 40;
#pragma unroll
                for (int t = 0; t < 8; ++t) {
                    ndw[off0 + t]      = c0[t];        // tile0
                    ndw[off0 + 16 + t] = c1[t];        // tile1 (+16 both sides)
                }
            }
            __builtin_amdgcn_wave_barrier();
            asm volatile("s_wait_dscnt 0x0" ::: "memory");

            const float num  = ndw[lane];
            const float den  = ndw[32 + lane];
            const float rden = __builtin_amdgcn_rcpf(den);

            orow[(size_t)i * CC + col] = num * rden * span + wmin_c;

#pragma unroll
            for (int k = 0; k < NBINS; ++k)
                mass[k] = fmaf(e[k], rden, mass[k]);   // bin_mass (sum over i)
        }
    }

    // --- entropy: total mass per block, then -sum p*log(p+eps) ---
    float s_local = 0.0f;
#pragma unroll
    for (int k = 0; k < NBINS; ++k) s_local += mass[k];
    red[tid] = s_local;
    __syncthreads();
    for (int s = 64; s > 0; s >>= 1) {
        if (tid < s) red[tid] += red[tid + s];
        __syncthreads();
    }
    const float rtot = 1.0f / (red[0] + EPSF);
    float ent = 0.0f;
#pragma unroll
    for (int k = 0; k < NBINS; ++k) {
        const float p = mass[k] * rtot;
        ent = fmaf(-p, __logf(p + EPSF), ent);
    }
    __syncthreads();
    red[tid] = ent;
    __syncthreads();
    for (int s = 64; s > 0; s >>= 1) {
        if (tid < s) red[tid] += red[tid + s];
        __syncthreads();
    }
    if (tid == 0) ws[n] = red[0];
}

__global__ __launch_bounds__(128)
void bq_entropy_sum(const float* __restrict__ ws, float* __restrict__ out_ent) {
    __shared__ float red[128];
    float s = 0.0f;
    for (int i = threadIdx.x; i < NB; i += 128) s += ws[i];
    red[threadIdx.x] = s;
    __syncthreads();
    for (int t = 64; t > 0; t >>= 1) {
        if (threadIdx.x < t) red[threadIdx.x] += red[threadIdx.x + t];
        __syncthreads();
    }
    if (threadIdx.x == 0) *out_ent = red[0];
}

extern "C" void kernel_launch(void* const* d_in, const int* in_sizes, int n_in,
                              void* d_out, int out_size, void* d_ws, size_t ws_size,
                              hipStream_t stream) {
    (void)in_sizes; (void)n_in; (void)out_size; (void)ws_size;
    const float* weight = (const float*)d_in[0];
    const float* wmin   = (const float*)d_in[1];
    const float* wmax   = (const float*)d_in[2];
    const float* bins   = (const float*)d_in[3];
    float* out = (float*)d_out;
    float* ws  = (float*)d_ws;

    hipLaunchKernelGGL(bq_main, dim3(NB), dim3(128), 0, stream,
                       weight, wmin, wmax, bins, out, ws);
    hipLaunchKernelGGL(bq_entropy_sum, dim3(1), dim3(128), 0, stream,
                       ws, out + (size_t)RR * CC);
}